// GATLayer_37838661878215
// MI455X (gfx1250) — compile-verified
//
#include <hip/hip_runtime.h>
#include <hip/hip_bf16.h>
#include <math.h>

typedef __attribute__((ext_vector_type(2))) float v2f;
typedef __attribute__((ext_vector_type(8))) float v8f;

#define NO 32
#define EO 64
#define DD 128   // D = NO + EO + NO, also OUT
#define TT 4
#define NODE_DIM 128
#define EDGE_DIM 64

__device__ __forceinline__ float leaky01(float x) { return x > 0.f ? x : 0.01f * x; }

// Float atomic max via sign-aware integer atomics (buffer must be init'ed to -inf).
__device__ __forceinline__ void atomicMaxF(float* addr, float val) {
  if (val >= 0.f) atomicMax((int*)addr, __float_as_int(val));
  else            atomicMin((unsigned int*)addr, __float_as_uint(val));
}

// ---------------------------------------------------------------------------
// Generic f32 WMMA GEMM: C[M,Ncols] = A[M,K] @ B[K,Ncols], row-major.
// One wave per 16x16 C tile; blockDim.x = 32 * (Ncols/16); grid.x = ceil(M/16).
// Uses V_WMMA_F32_16X16X4_F32 per CDNA5 ISA layouts:
//   A 16x4 : lane(l<16) VGPR0/1 = K0,K1 ; lane(l+16) = K2,K3 (row M = l)
//   B 4x16 : row-striped across lanes within a VGPR (mirrored halves)
//   C 16x16: VGPR v -> row v (lanes 0-15) / row v+8 (lanes 16-31)
// ---------------------------------------------------------------------------
__global__ void gemm_wmma_f32(const float* __restrict__ A, const float* __restrict__ B,
                              float* __restrict__ C, int M, int K, int Ncols) {
  const int wave = threadIdx.x >> 5;
  const int lane = threadIdx.x & 31;
  const int half = lane >> 4;    // 0 or 1
  const int l    = lane & 15;
  const int row0 = blockIdx.x * 16;
  const int col0 = wave * 16;
  if (col0 >= Ncols) return;     // uniform per wave

  const int rowA = min(row0 + l, M - 1);          // clamp tail rows; EXEC stays all-1
  const float* __restrict__ Arow = A + (size_t)rowA * K;

  v8f acc = {};
  for (int k0 = 0; k0 < K; k0 += 4) {
    const int ka = k0 + 2 * half;
    v2f a; a.x = Arow[ka]; a.y = Arow[ka + 1];
    v2f b; b.x = B[(size_t)ka * Ncols + col0 + l];
           b.y = B[(size_t)(ka + 1) * Ncols + col0 + l];
    acc = __builtin_amdgcn_wmma_f32_16x16x4_f32(false, a, false, b,
                                                (short)0, acc, false, false);
  }
#pragma unroll
  for (int v = 0; v < 8; ++v) {
    const int r = row0 + v + 8 * half;
    if (r < M) C[(size_t)r * Ncols + col0 + l] = acc[v];
  }
}

// ---------------------------------------------------------------------------
__global__ void fill_kernel(float* __restrict__ p, float v, int n) {
  int i = blockIdx.x * blockDim.x + threadIdx.x;
  if (i < n) p[i] = v;
}

// s1[e] = leaky( e_ft[e,:64]·w[0:64] + z[src[e],:32]·w[64:96] ); m1[src] = segmax
__global__ void score1_kernel(const float* __restrict__ e_ft, const float* __restrict__ z,
                              const int* __restrict__ src, const float* __restrict__ w_eattn,
                              float* __restrict__ s1, float* __restrict__ m1, int E) {
  int e = blockIdx.x * blockDim.x + threadIdx.x;
  if (e >= E) return;
  int s = src[e];
  const float4* ef = (const float4*)(e_ft + (size_t)e * EO);
  const float4* zs = (const float4*)(z + (size_t)s * NO);
  const float4* w0 = (const float4*)(w_eattn);
  const float4* w1 = (const float4*)(w_eattn + EO);
  float acc = 0.f;
#pragma unroll
  for (int i = 0; i < EO / 4; ++i) {
    float4 a = ef[i], b = w0[i];
    acc += a.x * b.x + a.y * b.y + a.z * b.z + a.w * b.w;
  }
#pragma unroll
  for (int i = 0; i < NO / 4; ++i) {
    float4 a = zs[i], b = w1[i];
    acc += a.x * b.x + a.y * b.y + a.z * b.z + a.w * b.w;
  }
  float sc = leaky01(acc);
  s1[e] = sc;
  atomicMaxF(&m1[s], sc);
}

// sbuf[e] <- exp(sbuf[e] - m[seg[e]]); den[seg] += exp
__global__ void expsum_kernel(const int* __restrict__ seg, float* __restrict__ sbuf,
                              const float* __restrict__ m, float* __restrict__ den, int E) {
  int e = blockIdx.x * blockDim.x + threadIdx.x;
  if (e >= E) return;
  int s = seg[e];
  float ex = __expf(sbuf[e] - m[s]);
  sbuf[e] = ex;
  atomicAdd(&den[s], ex);
}

// e_w[e,j] = (ex1[e]/den1[src[e]]) * e_ft[e,j]   (one thread per element)
__global__ void gamma_ew_kernel(const float* __restrict__ e_ft, const float* __restrict__ ex1,
                                const float* __restrict__ den1, const int* __restrict__ src,
                                float* __restrict__ e_w, long total) {
  long idx = (long)blockIdx.x * blockDim.x + threadIdx.x;
  if (idx >= total) return;
  int e = (int)(idx >> 6);  // /EO
  float gamma = ex1[e] / den1[src[e]];
  e_w[idx] = gamma * e_ft[idx];
}

// s2[e] = leaky( z_src·w[0:32] + e_w·w[32:96] + z_dst·w[96:128] ); m2[dst] = segmax
__global__ void score2_kernel(const float* __restrict__ z, const float* __restrict__ e_w,
                              const int* __restrict__ src, const int* __restrict__ dst,
                              const float* __restrict__ w_attn,
                              float* __restrict__ s2, float* __restrict__ m2, int E) {
  int e = blockIdx.x * blockDim.x + threadIdx.x;
  if (e >= E) return;
  int sv = src[e], dv = dst[e];
  const float4* zs = (const float4*)(z + (size_t)sv * NO);
  const float4* ew = (const float4*)(e_w + (size_t)e * EO);
  const float4* zd = (const float4*)(z + (size_t)dv * NO);
  const float4* wa = (const float4*)w_attn;
  float acc = 0.f;
#pragma unroll
  for (int i = 0; i < NO / 4; ++i) {
    float4 a = zs[i], b = wa[i];
    acc += a.x * b.x + a.y * b.y + a.z * b.z + a.w * b.w;
  }
#pragma unroll
  for (int i = 0; i < EO / 4; ++i) {
    float4 a = ew[i], b = wa[NO / 4 + i];
    acc += a.x * b.x + a.y * b.y + a.z * b.z + a.w * b.w;
  }
#pragma unroll
  for (int i = 0; i < NO / 4; ++i) {
    float4 a = zd[i], b = wa[(NO + EO) / 4 + i];
    acc += a.x * b.x + a.y * b.y + a.z * b.z + a.w * b.w;
  }
  float sc = leaky01(acc);
  s2[e] = sc;
  atomicMaxF(&m2[dv], sc);
}

// h[(dst*T+type), t] += alpha * msg[t]; msg regenerated on the fly (never materialized).
// 256 threads per block = 2 edges x 128 components.
__global__ void scatter_kernel(const float* __restrict__ z, const float* __restrict__ e_w,
                               const int* __restrict__ src, const int* __restrict__ dst,
                               const int* __restrict__ etype,
                               const float* __restrict__ ex2, const float* __restrict__ den2,
                               float* __restrict__ h, int E) {
  int e = blockIdx.x * 2 + (threadIdx.x >> 7);
  int t = threadIdx.x & 127;
  if (e >= E) return;
  int d = dst[e];
  float alpha = ex2[e] / den2[d];
  float msg;
  if (t < NO)            msg = z[(size_t)src[e] * NO + t];
  else if (t < NO + EO)  msg = e_w[(size_t)e * EO + (t - NO)];
  else                   msg = z[(size_t)d * NO + (t - NO - EO)];
  atomicAdd(&h[((size_t)d * TT + etype[e]) * DD + t], alpha * msg);
}

// zphi[t,d] = sum_n h[n,t,d]; block = 128 threads (one per d), strip-mined over rows.
__global__ void zphi_kernel(const float* __restrict__ h, float* __restrict__ zphi, int rows) {
  int d = threadIdx.x;
  int chunk = (rows + gridDim.x - 1) / gridDim.x;
  int r0 = blockIdx.x * chunk;
  int r1 = min(rows, r0 + chunk);
  float acc[TT] = {0.f, 0.f, 0.f, 0.f};
  for (int r = r0; r < r1; ++r) acc[r & 3] += h[(size_t)r * DD + d];  // r = n*4 + t
#pragma unroll
  for (int t = 0; t < TT; ++t) atomicAdd(&zphi[t * DD + d], acc[t]);
}

// beta = softmax(leaky(zphi @ w_sem)) over the 4 types. Single wave.
__global__ void beta_kernel(const float* __restrict__ zphi, const float* __restrict__ w_sem,
                            float* __restrict__ beta) {
  __shared__ float sv[TT];
  int t = threadIdx.x;
  if (t < TT) {
    float acc = 0.f;
    for (int d = 0; d < DD; ++d) acc += zphi[t * DD + d] * w_sem[d];
    sv[t] = leaky01(acc);
  }
  __syncthreads();
  if (t == 0) {
    float mx = sv[0];
    for (int i = 1; i < TT; ++i) mx = fmaxf(mx, sv[i]);
    float ex[TT], den = 0.f;
    for (int i = 0; i < TT; ++i) { ex[i] = __expf(sv[i] - mx); den += ex[i]; }
    for (int i = 0; i < TT; ++i) beta[i] = ex[i] / den;
  }
}

// hb[n,d] = sum_t beta[t] * h[n,t,d]
__global__ void hb_kernel(const float* __restrict__ h, const float* __restrict__ beta,
                          float* __restrict__ hb, long total) {
  long idx = (long)blockIdx.x * blockDim.x + threadIdx.x;
  if (idx >= total) return;
  long node = idx >> 7;
  int d = (int)(idx & 127);
  const float* hr = h + ((size_t)node * TT) * DD + d;
  hb[idx] = beta[0] * hr[0] + beta[1] * hr[DD] + beta[2] * hr[2 * DD] + beta[3] * hr[3 * DD];
}

// ---------------------------------------------------------------------------
extern "C" void kernel_launch(void* const* d_in, const int* in_sizes, int n_in,
                              void* d_out, int out_size, void* d_ws, size_t ws_size,
                              hipStream_t stream) {
  (void)n_in; (void)out_size; (void)ws_size;
  const float* node_features = (const float*)d_in[0];
  const float* edge_features = (const float*)d_in[1];
  const int*   src           = (const int*)d_in[2];
  const int*   dst           = (const int*)d_in[3];
  const int*   etype         = (const int*)d_in[4];
  const float* W_n           = (const float*)d_in[5];
  const float* W_e           = (const float*)d_in[6];
  const float* w_eattn       = (const float*)d_in[7];
  const float* w_attn        = (const float*)d_in[8];
  const float* w_sem         = (const float*)d_in[9];
  const float* W_fc2         = (const float*)d_in[10];

  const int n = in_sizes[0] / NODE_DIM;   // 25000
  const int E = in_sizes[1] / EDGE_DIM;   // 400000

  float* Z   = (float*)d_out;                    // [n, 128]
  float* e_w = Z + (size_t)n * DD;               // [E, 64] (second output)

  // ---- workspace carve-out (16B aligned slots) ----
  float* ws = (float*)d_ws;
  size_t cur = 0;
  auto alloc = [&](size_t cnt) { float* p = ws + cur; cur += (cnt + 3) & ~(size_t)3; return p; };
  float* z    = alloc((size_t)n * NO);
  float* eft  = alloc((size_t)E * EO);
  float* s1   = alloc((size_t)E);
  float* m1   = alloc((size_t)n);
  float* den1 = alloc((size_t)n);
  float* s2   = alloc((size_t)E);
  float* m2   = alloc((size_t)n);
  float* den2 = alloc((size_t)n);
  float* h    = alloc((size_t)n * TT * DD);
  float* zphi = alloc((size_t)TT * DD);
  float* beta = alloc((size_t)TT);
  float* hb   = alloc((size_t)n * DD);

  // ---- init accumulators (every call; graph-capture safe) ----
  hipMemsetAsync(den1, 0, (size_t)n * sizeof(float), stream);
  hipMemsetAsync(den2, 0, (size_t)n * sizeof(float), stream);
  hipMemsetAsync(h,    0, (size_t)n * TT * DD * sizeof(float), stream);
  hipMemsetAsync(zphi, 0, (size_t)TT * DD * sizeof(float), stream);
  fill_kernel<<<(n + 255) / 256, 256, 0, stream>>>(m1, -INFINITY, n);
  fill_kernel<<<(n + 255) / 256, 256, 0, stream>>>(m2, -INFINITY, n);

  // ---- projections (WMMA f32 16x16x4) ----
  gemm_wmma_f32<<<dim3((n + 15) / 16), dim3(32 * (NO / 16)), 0, stream>>>(
      node_features, W_n, z, n, NODE_DIM, NO);
  gemm_wmma_f32<<<dim3((E + 15) / 16), dim3(32 * (EO / 16)), 0, stream>>>(
      edge_features, W_e, eft, E, EDGE_DIM, EO);

  // ---- edge-level softmax over src segments -> e_w ----
  score1_kernel<<<(E + 255) / 256, 256, 0, stream>>>(eft, z, src, w_eattn, s1, m1, E);
  expsum_kernel<<<(E + 255) / 256, 256, 0, stream>>>(src, s1, m1, den1, E);
  {
    long total = (long)E * EO;
    gamma_ew_kernel<<<(unsigned)((total + 255) / 256), 256, 0, stream>>>(
        eft, s1, den1, src, e_w, total);
  }

  // ---- GAT attention softmax over dst segments, type-grouped scatter ----
  score2_kernel<<<(E + 255) / 256, 256, 0, stream>>>(z, e_w, src, dst, w_attn, s2, m2, E);
  expsum_kernel<<<(E + 255) / 256, 256, 0, stream>>>(dst, s2, m2, den2, E);
  scatter_kernel<<<(E + 1) / 2, 256, 0, stream>>>(z, e_w, src, dst, etype, s2, den2, h, E);

  // ---- semantic attention + output GEMM ----
  zphi_kernel<<<128, 128, 0, stream>>>(h, zphi, n * TT);
  beta_kernel<<<1, 32, 0, stream>>>(zphi, w_sem, beta);
  {
    long total = (long)n * DD;
    hb_kernel<<<(unsigned)((total + 255) / 256), 256, 0, stream>>>(h, beta, hb, total);
  }
  gemm_wmma_f32<<<dim3((n + 15) / 16), dim3(32 * (DD / 16)), 0, stream>>>(
      hb, W_fc2, Z, n, DD, DD);
}